// FloodFill_7507602833763
// MI455X (gfx1250) — compile-verified
//
#include <hip/hip_runtime.h>
#include <stdint.h>

namespace {
constexpr int BATCH = 64;
constexpr int CIN = 4;
constexpr int H = 96, W = 96, HW = H * W;
constexpr int MAX_ITERS = 129;
constexpr int WALL_CHAN = 1, SRC_CHAN = 3;
constexpr int CY = H / 2, CX = W / 2;
constexpr int TY = 3;                // blockDim.y
constexpr int RPT = H / TY;          // 32 rows per thread
constexpr int NTHREADS = W * TY;     // 288 threads = 9 wave32s
}

// Low 32 bits of a flat shared-aperture pointer = LDS byte address (gfx1250
// aperture: addr[31:0] is the LDS offset).
__device__ __forceinline__ uint32_t lds_addr32(const void* p) {
  return (uint32_t)(uintptr_t)p;
}

// gfx1250 async DMA: 16 bytes per lane, global -> LDS, tracked by ASYNCcnt.
__device__ __forceinline__ void async_g2l_b128(uint32_t lds_byte, const void* g) {
  asm volatile("global_load_async_to_lds_b128 %0, %1, off"
               :: "v"(lds_byte), "v"(g)
               : "memory");
}

__device__ __forceinline__ void wait_async_zero() {
  asm volatile("s_wait_asynccnt 0x0" ::: "memory");
}

template <bool PASS2>
__global__ __launch_bounds__(NTHREADS) void flood_kernel(
    const float* __restrict__ in, float* __restrict__ out,
    int* __restrict__ ws) {
  __shared__ float fld[HW];   // 36 KB: flood grid, updated in place
  __shared__ int ctrl_s;      // pass1: crossing step (0 = not yet)
  __shared__ float actb[9];   // pass2: 3x3 age neighborhood of center

  const int x = threadIdx.x;        // 0..95 (stride-1 across lanes)
  const int ty = threadIdx.y;       // 0..2
  const int tid = ty * W + x;       // 0..287
  const int y0 = ty * RPT;
  const int b = blockIdx.x;

  const float* gin = in + (size_t)b * CIN * HW;
  const float* wallg = gin + WALL_CHAN * HW;
  const float* srcg = gin + SRC_CHAN * HW;

  // ---- Stage initial flood into LDS via async global->LDS DMA ----
  {
    const uint32_t dst = lds_addr32(fld);
    #pragma unroll
    for (int k = 0; k < (HW / 4) / NTHREADS; ++k) {   // 8 wave-level b128 DMAs
      const int i = tid + k * NTHREADS;
      async_g2l_b128(dst + (uint32_t)i * 16u, srcg + i * 4);
    }
  }

  // Wall stays in VGPRs for the whole run (read every iteration).
  float wallr[RPT];
  #pragma unroll
  for (int r = 0; r < RPT; ++r) wallr[r] = wallg[(y0 + r) * W + x];

  float ager[RPT];
  if (PASS2) {
    #pragma unroll
    for (int r = 0; r < RPT; ++r) ager[r] = 0.0f;
  }

  if (tid == 0) ctrl_s = 0;

  wait_async_zero();
  __syncthreads();

  int T = MAX_ITERS;
  if (PASS2) {
    const int mt = ws[0];            // max_b t_b from pass 1
    T = mt < MAX_ITERS ? mt : MAX_ITERS;
  }

  const bool isCenter = (x == CX) && (ty == CY / RPT);
  float ageC = 0.0f;
  bool crossed = false;

  const int yup0 = (y0 == 0) ? H - 1 : y0 - 1;
  const int xl = (x == 0) ? W - 1 : x - 1;
  const int xr = (x == W - 1) ? 0 : x + 1;

  #pragma unroll 1
  for (int it = 0; it < T; ++it) {
    float nf[RPT];
    float centerOld = 0.0f;
    // Phase A: compute new flood into registers (reads only, conflict-free).
    float up = fld[yup0 * W + x];
    float mid = fld[y0 * W + x];
    #pragma unroll
    for (int r = 0; r < RPT; ++r) {
      const int y = y0 + r;
      const int yd = (y == H - 1) ? 0 : y + 1;
      const float dn = fld[yd * W + x];
      const float lf = fld[y * W + xl];
      const float rt = fld[y * W + xr];
      float s = up + mid + dn + lf + rt - 6.0f * wallr[r];
      s = fminf(fmaxf(s, 0.0f), 1.0f);
      nf[r] = s;
      if (PASS2) ager[r] += mid;                 // age += old flood
      if (!PASS2 && r == (CY % RPT)) centerOld = mid;
      up = mid;
      mid = dn;
    }
    if (!PASS2 && isCenter) {
      ageC += centerOld;                          // center age after step it+1
      if (!crossed && ageC > 0.1f) {
        crossed = true;
        ctrl_s = it + 1;                          // written exactly once
      }
    }
    __syncthreads();                              // flag write -> flag read
    if (!PASS2) {
      if (ctrl_s != 0) break;                     // uniform exit: no deadlock
    }
    // Phase B: write back in place.
    #pragma unroll
    for (int r = 0; r < RPT; ++r) fld[(y0 + r) * W + x] = nf[r];
    __syncthreads();                              // writes -> next reads
  }

  if (!PASS2) {
    if (tid == 0) {
      int t = ctrl_s;
      if (t == 0) t = MAX_ITERS + 1;              // never crossed -> clamp later
      atomicMax(ws, t);
    }
    return;
  }

  // ---- Pass 2 outputs: state = [input, flood, age], plus action ----
  float* gout = out + (size_t)b * 6 * HW;

  // Channels 0..3: straight copy, 16B vectors (bases are 16B aligned).
  const float4* gi4 = (const float4*)gin;
  float4* go4 = (float4*)gout;
  #pragma unroll
  for (int k = 0; k < (CIN * HW / 4) / NTHREADS; ++k) {
    const int i = tid + k * NTHREADS;
    go4[i] = gi4[i];
  }

  float* gflood = gout + 4 * HW;
  float* gage = gout + 5 * HW;
  #pragma unroll
  for (int r = 0; r < RPT; ++r) {
    const int y = y0 + r;
    gflood[y * W + x] = fld[y * W + x];
    gage[y * W + x] = ager[r];
  }

  // Gather the 3x3 age block around (CY, CX) from register files.
  if (x >= CX - 1 && x <= CX + 1 && ty == CY / RPT) {
    #pragma unroll
    for (int r = 0; r < RPT; ++r) {
      const int y = y0 + r;
      if (y >= CY - 1 && y <= CY + 1)
        actb[(y - (CY - 1)) * 3 + (x - (CX - 1))] = ager[r];
    }
  }
  __syncthreads();

  if (tid == 0) {
    const float mask[9] = {0.f, 1.f, 0.f, 1.f, 1.f, 1.f, 0.f, 1.f, 0.f};
    float best = actb[0] * mask[0];
    int bi = 0;
    #pragma unroll
    for (int i = 1; i < 9; ++i) {
      const float v = actb[i] * mask[i];
      if (v > best) { best = v; bi = i; }         // strict '>' = first-max, like jnp.argmax
    }
    int col = bi % 3, row = bi / 3;
    if (col == 0) col = 1;
    if (row == 0) row = 1;
    const int table[3][3] = {{0, 0, 0}, {1, 2, 3}, {0, 4, 0}};
    out[(size_t)BATCH * 6 * HW + b] = (float)table[col][row];
  }
}

__global__ void ws_init_kernel(int* ws) { ws[0] = 0; }

extern "C" void kernel_launch(void* const* d_in, const int* in_sizes, int n_in,
                              void* d_out, int out_size, void* d_ws, size_t ws_size,
                              hipStream_t stream) {
  (void)in_sizes; (void)n_in; (void)out_size; (void)ws_size;
  const float* in = (const float*)d_in[0];
  float* out = (float*)d_out;
  int* ws = (int*)d_ws;

  dim3 block(W, TY, 1);
  ws_init_kernel<<<1, 1, 0, stream>>>(ws);
  // Pass 1: per-image crossing step t_b, reduced with atomicMax.
  flood_kernel<false><<<BATCH, block, 0, stream>>>(in, out, ws);
  // Pass 2: run exactly T = min(max t_b, 129) iterations and emit outputs.
  flood_kernel<true><<<BATCH, block, 0, stream>>>(in, out, ws);
}